// NLinear_60954175865084
// MI455X (gfx1250) — compile-verified
//
#include <hip/hip_runtime.h>

typedef __attribute__((ext_vector_type(2))) float v2f;
typedef __attribute__((ext_vector_type(8))) float v8f;

#define NST       300
#define BATCH     16384
#define IN_DIM    352
#define Z_DIM     288
#define D_MODEL   64
#define OUT_DIM   72
#define A_STRIDE  353          // pad: 353 % 64 coprime-ish -> conflict-free column reads
#define MAX_TILES 32           // up to 512 samples/station (mean ~55; overflow prob ~0)

// ---------------- workspace layout (int32) ----------------
// [0,300)      counts
// [300,600)    cursor
// [600,901)    offsets (exclusive prefix, offsets[300]=B)
// [1024,17408) perm
#define WS_COUNTS  0
#define WS_CURSOR  300
#define WS_OFFSETS 600
#define WS_PERM    1024

__global__ void nl_init(int* ws) {
    int i = blockIdx.x * blockDim.x + threadIdx.x;
    if (i < 600) ws[i] = 0;           // counts + cursor
}

__global__ void nl_count(const int* __restrict__ stations, int* __restrict__ counts) {
    int i = blockIdx.x * blockDim.x + threadIdx.x;
    if (i < BATCH) atomicAdd(&counts[stations[i]], 1);
}

__global__ void nl_scan(const int* __restrict__ counts, int* __restrict__ offsets) {
    __shared__ int tmp[512];
    int t = threadIdx.x;
    tmp[t] = (t < NST) ? counts[t] : 0;
    __syncthreads();
    for (int d = 1; d < 512; d <<= 1) {
        int v = (t >= d) ? tmp[t - d] : 0;
        __syncthreads();
        tmp[t] += v;
        __syncthreads();
    }
    if (t <= NST) offsets[t] = (t == 0) ? 0 : tmp[t - 1];
}

__global__ void nl_scatter(const int* __restrict__ stations,
                           const int* __restrict__ offsets,
                           int* __restrict__ cursor,
                           int* __restrict__ perm) {
    int i = blockIdx.x * blockDim.x + threadIdx.x;
    if (i < BATCH) {
        int st  = stations[i];
        int pos = offsets[st] + atomicAdd(&cursor[st], 1);
        perm[pos] = i;
    }
}

// ---------------- main fused gather + GEMM (WMMA f32 16x16x4) ----------------
// grid.x = station, grid.y = M-tile; 160 threads = 5 wave32s, wave w -> N cols [16w,16w+16)
__global__ __launch_bounds__(160)
void nl_gemm(const float* __restrict__ z,
             const int*   __restrict__ weekday,
             const int*   __restrict__ timeind,
             const float* __restrict__ W,
             const float* __restrict__ bias,
             const float* __restrict__ week_tab,
             const float* __restrict__ time_tab,
             const float* __restrict__ x_loc,
             const float* __restrict__ x_scale,
             const int*   __restrict__ offsets,
             const int*   __restrict__ perm,
             float*       __restrict__ out) {
    __shared__ float Alds[16 * A_STRIDE];
    __shared__ int   smp[16];

    const int st    = blockIdx.x;
    const int start = offsets[st];
    const int cnt   = offsets[st + 1] - start;
    const int tbase = blockIdx.y * 16;
    if (tbase >= cnt) return;                 // block-uniform exit

    const int tid = threadIdx.x;
    if (tid < 16) {
        int off  = (tbase + tid < cnt) ? tid : 0;   // duplicate row 0 for padding rows
        smp[tid] = perm[start + tbase + off];
    }
    __syncthreads();

    const float loc0 = x_loc[st * 2 + 0],   loc1 = x_loc[st * 2 + 1];
    const float sc0  = x_scale[st * 2 + 0], sc1  = x_scale[st * 2 + 1];

    // Build A tile (16 x 352) in LDS: normalized z ++ (week+time embedding)
    for (int idx = tid; idx < 16 * IN_DIM; idx += 160) {
        int m = idx / IN_DIM;
        int k = idx - m * IN_DIM;
        int bs = smp[m];
        float v;
        if (k < Z_DIM) {
            float zz = z[(size_t)bs * Z_DIM + k];
            v = (k & 1) ? (zz - loc1) / sc1 : (zz - loc0) / sc0;
        } else {
            int c  = k - Z_DIM;
            int wd = weekday[bs];
            int ti = timeind[bs];
            v = week_tab[((size_t)st * 7 + wd) * D_MODEL + c]
              + time_tab[((size_t)st * 96 + ti) * D_MODEL + c];
        }
        Alds[m * A_STRIDE + k] = v;
    }
    __syncthreads();

    const int lane = tid & 31;
    const int wave = tid >> 5;
    const int n0   = wave * 16;
    const int col  = n0 + (lane & 15);
    const int colC = (col < OUT_DIM) ? col : (OUT_DIM - 1);   // clamp OOB B loads
    const int kh   = (lane >> 4) << 1;                        // 0 (lo half) / 2 (hi half)
    const int mRow = lane & 15;

    const float* Wst  = W + (size_t)st * IN_DIM * OUT_DIM;
    const float* Arow = &Alds[mRow * A_STRIDE];

    v8f acc = {};
    #pragma unroll 4
    for (int k0 = 0; k0 < IN_DIM; k0 += 4) {
        v2f a, bm;
        a.x  = Arow[k0 + kh];
        a.y  = Arow[k0 + kh + 1];
        bm.x = Wst[(size_t)(k0 + kh) * OUT_DIM + colC];
        bm.y = Wst[(size_t)(k0 + kh + 1) * OUT_DIM + colC];
        // D = A(16x4,f32) * B(4x16,f32) + C  -> v_wmma_f32_16x16x4_f32
        acc = __builtin_amdgcn_wmma_f32_16x16x4_f32(
                  false, a, false, bm, (short)0, acc, false, false);
    }

    if (col < OUT_DIM) {
        const float bv   = bias[st * OUT_DIM + col];
        const float scf  = (col & 1) ? sc1 : sc0;
        const float locf = (col & 1) ? loc1 : loc0;
        const int mhalf  = (lane >> 4) * 8;     // D: VGPR r -> M=r (lo lanes) / r+8 (hi lanes)
        #pragma unroll
        for (int r = 0; r < 8; ++r) {
            int M = mhalf + r;
            if (tbase + M < cnt) {
                int bs = smp[M];
                out[(size_t)bs * OUT_DIM + col] = (acc[r] + bv) * scf + locf;
            }
        }
    }
}

extern "C" void kernel_launch(void* const* d_in, const int* in_sizes, int n_in,
                              void* d_out, int out_size, void* d_ws, size_t ws_size,
                              hipStream_t stream) {
    const float* z        = (const float*)d_in[0];
    const int*   stations = (const int*)  d_in[1];
    const int*   weekday  = (const int*)  d_in[2];
    const int*   timeind  = (const int*)  d_in[3];
    const float* W        = (const float*)d_in[4];
    const float* b        = (const float*)d_in[5];
    const float* week_tab = (const float*)d_in[6];
    const float* time_tab = (const float*)d_in[7];
    const float* x_loc    = (const float*)d_in[8];
    const float* x_scale  = (const float*)d_in[9];
    float* out = (float*)d_out;
    int*   ws  = (int*)d_ws;

    int* counts  = ws + WS_COUNTS;
    int* cursor  = ws + WS_CURSOR;
    int* offsets = ws + WS_OFFSETS;
    int* perm    = ws + WS_PERM;

    nl_init   <<<1, 640, 0, stream>>>(ws);
    nl_count  <<<(BATCH + 255) / 256, 256, 0, stream>>>(stations, counts);
    nl_scan   <<<1, 512, 0, stream>>>(counts, offsets);
    nl_scatter<<<(BATCH + 255) / 256, 256, 0, stream>>>(stations, offsets, cursor, perm);
    nl_gemm   <<<dim3(NST, MAX_TILES), 160, 0, stream>>>(
        z, weekday, timeind, W, b, week_tab, time_tab, x_loc, x_scale,
        offsets, perm, out);
}